// PostProcess_75411035783272
// MI455X (gfx1250) — compile-verified
//
#include <hip/hip_runtime.h>
#include <stdint.h>

// ---------------- problem constants ----------------
#define NB    256      // batch
#define NQ    1000     // queries
#define NC    81       // classes
#define QC    81000    // NQ*NC elements per batch
#define KTOP  100
#define NBINS 32768    // histogram over top 16 bits of positive float
#define TILE  4096     // floats per staged tile (16 KB)
#define NTILES 20      // ceil(81000/4096)
#define CAP   2048     // candidate buffer slots
#define NT    1024     // threads per block (32 wave32s)

// ---------------- dynamic LDS layout (bytes) ----------------
#define OFF_CAND   0            // 2048 * 8   = 16384
#define OFF_HIST   16384        // 32768 * 4  = 131072
#define OFF_STAGE  147456       // 2 * 4096*4 = 32768
#define OFF_LA     180224       // 1024 * 4
#define OFF_LB     184320       // 1024 * 4
#define OFF_SUMS   188416       // 1024 * 4
#define OFF_MISC   192512       // scalars
#define SMEM_BYTES 192544

// ---------------- CDNA5 async global->LDS (ASYNCcnt path) ----------------
__device__ __forceinline__ void async_load16(uint32_t lds_off, const void* gaddr) {
  // VDST = LDS byte address (per lane), VADDR = 64-bit global address (per lane)
  asm volatile("global_load_async_to_lds_b128 %0, %1, off"
               :: "v"(lds_off), "v"(gaddr)
               : "memory");
}
__device__ __forceinline__ void wait_async_le1() {
  asm volatile("s_wait_asynccnt 1" ::: "memory");
}
__device__ __forceinline__ void wait_async_0() {
  asm volatile("s_wait_asynccnt 0" ::: "memory");
}

// One streaming pass over a batch's logits. Each thread t consumes exactly the
// 16 bytes (4 floats) its own lane DMA'd, so tile consumption only needs the
// wave-local s_wait_asynccnt (async loads complete in order).
template <bool PASS2>
__device__ __forceinline__ void stream_pass(const float* __restrict__ logitsB,
                                            const float* __restrict__ LA,
                                            const float* __restrict__ LB,
                                            uint32_t* __restrict__ hist,
                                            uint64_t* __restrict__ cand,
                                            uint32_t* __restrict__ misc,
                                            const float* __restrict__ stage,
                                            uint32_t stage_lds, uint32_t cut, int t) {
  {
    int base = 4 * t;                       // tile 0
    if (base > QC - 4) base = QC - 4;       // clamp tail lanes (16B aligned)
    async_load16(stage_lds + 16u * (uint32_t)t, logitsB + base);
  }
  for (int k = 0; k < NTILES; ++k) {
    if (k + 1 < NTILES) {
      int base = (k + 1) * TILE + 4 * t;
      if (base > QC - 4) base = QC - 4;
      async_load16(stage_lds + (uint32_t)((k + 1) & 1) * (TILE * 4) + 16u * (uint32_t)t,
                   logitsB + base);
      wait_async_le1();                     // tile k landed (in-order completion)
    } else {
      wait_async_0();
    }
    const float* buf = stage + (k & 1) * TILE;
    const int ebase = k * TILE + 4 * t;
#pragma unroll
    for (int u = 0; u < 4; ++u) {
      int e = ebase + u;
      if (e >= QC) break;
      int q = e / NC;
      int c = e - q * NC;
      if (c >= 60 && c < 80) continue;      // INVALID classes -> prob 0, skip
      float p;
      if (c == 80) {
        p = LB[q];                          // obj_prob * unk
      } else {
        float x = buf[4 * t + u];
        p = LA[q] * (1.0f / (1.0f + __expf(-x)));  // obj*(1-unk)*sigmoid(logit)
      }
      uint32_t bits = __float_as_uint(p);   // p >= 0 -> bits monotonic
      if (bits == 0u) continue;
      if (!PASS2) {
        atomicAdd(&hist[bits >> 16], 1u);   // ds_add on LDS
      } else if ((bits >> 16) >= cut) {
        uint32_t pos = atomicAdd(&misc[0], 1u);
        if (pos < CAP)
          cand[pos] = ((uint64_t)bits << 32) | (uint32_t)(~(uint32_t)e);
      }
    }
  }
}

__global__ __launch_bounds__(NT, 1)
void postprocess_topk_kernel(const float* __restrict__ logits,
                             const float* __restrict__ obj,
                             const float* __restrict__ pboxes,
                             const float* __restrict__ unk,
                             const float* __restrict__ tsz,
                             float* __restrict__ out) {
  extern __shared__ unsigned char smem[];
  uint64_t* cand = (uint64_t*)(smem + OFF_CAND);
  uint32_t* hist = (uint32_t*)(smem + OFF_HIST);
  float*    stage = (float*)(smem + OFF_STAGE);
  float*    LA = (float*)(smem + OFF_LA);
  float*    LB = (float*)(smem + OFF_LB);
  uint32_t* sums = (uint32_t*)(smem + OFF_SUMS);
  uint32_t* misc = (uint32_t*)(smem + OFF_MISC);

  const int t = threadIdx.x;
  const int b = blockIdx.x;
  const float* logitsB = logits + (size_t)b * QC;
  // flat LDS aperture: low 32 bits of a generic pointer == LDS byte offset
  const uint32_t stage_lds = (uint32_t)(uintptr_t)(const void*)stage;

  // ---- init LDS ----
  for (int i = t; i < NBINS; i += NT) hist[i] = 0u;
  for (int i = t; i < CAP; i += NT) cand[i] = 0ull;
  if (t == 0) { misc[0] = 0u; misc[1] = 0u; }
  if (t < NQ) {
    float eo = __expf(-obj[(size_t)b * NQ + t]);            // exp(-pred_obj)
    float us = 1.0f / (1.0f + __expf(-unk[(size_t)b * NQ + t]));
    LA[t] = eo * (1.0f - us);
    LB[t] = eo * us;
  }
  __syncthreads();

  // ---- pass 1: histogram of score bits ----
  stream_pass<false>(logitsB, LA, LB, hist, cand, misc, stage, stage_lds, 0u, t);
  __syncthreads();

  // ---- per-thread chunk sums (32 bins each) ----
  {
    uint32_t s = 0;
    const int base = t * (NBINS / NT);
#pragma unroll
    for (int i = 0; i < NBINS / NT; ++i) s += hist[base + i];
    sums[t] = s;
  }
  __syncthreads();
  // inclusive suffix scan over chunk sums
  for (int off = 1; off < NT; off <<= 1) {
    uint32_t v = (t + off < NT) ? sums[t + off] : 0u;
    __syncthreads();
    sums[t] += v;
    __syncthreads();
  }
  // locate the bin containing the K-th largest value
  {
    uint32_t Sthis = sums[t];
    uint32_t Snext = (t + 1 < NT) ? sums[t + 1] : 0u;
    if (t == 0 && Sthis < KTOP) misc[1] = 0u;     // fewer than K positives
    if (Sthis >= KTOP && Snext < KTOP) {
      uint32_t acc = Snext;
      const int base = t * (NBINS / NT);
      uint32_t cutv = (uint32_t)base;
      for (int i = NBINS / NT - 1; i >= 0; --i) {
        acc += hist[base + i];
        if (acc >= KTOP) { cutv = (uint32_t)(base + i); break; }
      }
      misc[1] = cutv;
    }
  }
  __syncthreads();
  const uint32_t cut = misc[1];

  // ---- pass 2: collect candidates >= cut bin ----
  stream_pass<true>(logitsB, LA, LB, hist, cand, misc, stage, stage_lds, cut, t);
  __syncthreads();

  // ---- bitonic sort CAP keys descending; key = (bits<<32)|~idx (ties -> low idx) ----
  for (unsigned k = 2; k <= CAP; k <<= 1) {
    for (unsigned j = k >> 1; j > 0; j >>= 1) {
      for (unsigned i = (unsigned)t; i < CAP; i += NT) {
        unsigned l = i ^ j;
        if (l > i) {
          uint64_t a = cand[i], c2 = cand[l];
          bool descBlk = ((i & k) == 0);
          if (descBlk ? (a < c2) : (a > c2)) { cand[i] = c2; cand[l] = a; }
        }
      }
      __syncthreads();
    }
  }

  // ---- emit top-100: scores | labels | boxes (flat, in return order) ----
  if (t < KTOP) {
    uint64_t key = cand[t];
    uint32_t sbits = (uint32_t)(key >> 32);
    float score = 0.f, bx0 = 0.f, by0 = 0.f, bx1 = 0.f, by1 = 0.f;
    int label = 0;
    if (sbits != 0u) {
      uint32_t idx = ~(uint32_t)key;
      int q = (int)(idx / NC);
      label = (int)(idx - (uint32_t)q * NC);
      score = __uint_as_float(sbits);
      const float* bp = pboxes + ((size_t)b * NQ + q) * 4;
      float cx = bp[0], cy = bp[1], w = bp[2], h = bp[3];
      float ih = tsz[b * 2 + 0], iw = tsz[b * 2 + 1];
      bx0 = (cx - 0.5f * w) * iw;
      by0 = (cy - 0.5f * h) * ih;
      bx1 = (cx + 0.5f * w) * iw;
      by1 = (cy + 0.5f * h) * ih;
    }
    const int o = b * KTOP + t;
    out[o] = score;                                  // scores
    out[NB * KTOP + o] = (float)label;               // labels
    float* ob = out + 2 * NB * KTOP + (size_t)o * 4; // boxes
    ob[0] = bx0; ob[1] = by0; ob[2] = bx1; ob[3] = by1;
  }
}

extern "C" void kernel_launch(void* const* d_in, const int* in_sizes, int n_in,
                              void* d_out, int out_size, void* d_ws, size_t ws_size,
                              hipStream_t stream) {
  (void)in_sizes; (void)n_in; (void)out_size; (void)d_ws; (void)ws_size;
  const float* logits = (const float*)d_in[0];  // (256,1000,81)
  const float* obj    = (const float*)d_in[1];  // (256,1000)
  const float* boxes  = (const float*)d_in[2];  // (256,1000,4)
  const float* unk    = (const float*)d_in[3];  // (256,1000)
  const float* tsz    = (const float*)d_in[4];  // (256,2)
  float* out = (float*)d_out;                   // 25600 + 25600 + 102400 floats

  static bool attr_set = false;  // attribute config, not work/state: deterministic
  if (!attr_set) {
    hipFuncSetAttribute((const void*)postprocess_topk_kernel,
                        hipFuncAttributeMaxDynamicSharedMemorySize, SMEM_BYTES);
    attr_set = true;
  }
  hipLaunchKernelGGL(postprocess_topk_kernel, dim3(NB), dim3(NT), SMEM_BYTES, stream,
                     logits, obj, boxes, unk, tsz, out);
}